// Loss_20495583936604
// MI455X (gfx1250) — compile-verified
//
#include <hip/hip_runtime.h>
#include <hip/hip_bf16.h>

#define SEGS 128
#define CHUNK 1024   // floats per LDS staging buffer (4 KB each)

typedef __attribute__((ext_vector_type(2))) float v2f;
typedef __attribute__((ext_vector_type(8))) float v8f;

// ---------------------------------------------------------------------------
// Kernel 1: deterministic counting-sort of scores into per-segment pos/neg
// buckets. One thread per (segment,label) pair (128*2 = 256 threads, 1 block).
// Inputs are tiny (96KB) and broadcast-read, so the serial scans are cheap and
// fully deterministic (no atomics -> bit-stable bucket order across replays).
// ---------------------------------------------------------------------------
__global__ __launch_bounds__(256) void bucket_kernel(
    const int* __restrict__ b, const float* __restrict__ s,
    const int* __restrict__ y, int n,
    unsigned* __restrict__ posCnt, unsigned* __restrict__ negCnt,
    unsigned* __restrict__ posOff, unsigned* __restrict__ negOff,
    float* __restrict__ posScores, float* __restrict__ negScores)
{
    __shared__ unsigned cnt[2 * SEGS];
    __shared__ unsigned offs[2 * SEGS];

    const int t   = threadIdx.x;       // 0..255
    const int g   = t >> 1;            // segment
    const int lab = t & 1;             // 0 = neg, 1 = pos

    // pass 1: count
    unsigned c = 0;
    for (int i = 0; i < n; ++i)
        if (b[i] == g && y[i] == lab) ++c;
    cnt[t] = c;
    __syncthreads();

    // exclusive prefix per label region (128 entries each, serial: trivial)
    if (t == 0) {
        unsigned pacc = 0, nacc = 0;
        for (int gg = 0; gg < SEGS; ++gg) {
            offs[2 * gg + 1] = pacc; pacc += cnt[2 * gg + 1];
            offs[2 * gg + 0] = nacc; nacc += cnt[2 * gg + 0];
        }
    }
    __syncthreads();

    // pass 2: stable scatter into compacted per-segment arrays
    const unsigned o = offs[t];
    float* dst = lab ? posScores : negScores;
    unsigned k = o;
    for (int i = 0; i < n; ++i)
        if (b[i] == g && y[i] == lab) dst[k++] = s[i];

    if (lab) { posCnt[g] = c; posOff[g] = o; }
    else     { negCnt[g] = c; negOff[g] = o; }
}

// ---------------------------------------------------------------------------
// Kernel 2: one block per segment. Stage pos/neg scores in LDS, then each
// wave32 consumes 16x16 (pos x neg) tiles. The outer difference
// d[m,n] = s_neg[n] - s_pos[m] is computed by V_WMMA_F32_16X16X4_F32 as a
// rank-2 product: A = [-s_pos | 1 | 0 | 0] (16x4), B = [1 ; s_neg ; 0 ; 0]
// (4x16), C = 0. Bit-identical to scalar f32 subtraction (all products
// exact); the matrix pipe builds the tile while the VALU runs only the
// softplus transcendentals (v_exp_f32 / v_log_f32), which are the true
// throughput limit. The raw-ISA softplus is safe because the log argument
// 1 + exp(-|x|) is always in (1, 2] -- no denorm/overflow guards needed.
// ---------------------------------------------------------------------------
__global__ __launch_bounds__(256) void pair_loss_kernel(
    const float* __restrict__ posScores, const float* __restrict__ negScores,
    const unsigned* __restrict__ posCnt, const unsigned* __restrict__ negCnt,
    const unsigned* __restrict__ posOff, const unsigned* __restrict__ negOff,
    float* __restrict__ partial)
{
    __shared__ float ldsP[CHUNK];
    __shared__ float ldsN[CHUNK];
    __shared__ float waveSum[8];

    const int g  = blockIdx.x;
    const int np = (int)posCnt[g];
    const int nn = (int)negCnt[g];
    const float* P = posScores + posOff[g];
    const float* Q = negScores + negOff[g];

    const int tid  = threadIdx.x;
    const int lane = tid & 31;
    const int wave = tid >> 5;     // 8 wave32 waves per block
    const bool lo  = lane < 16;

    const float LOG2E = 1.44269504088896340736f;
    const float LN2   = 0.69314718055994530942f;

    float acc = 0.0f;

    for (int pc = 0; pc < np; pc += CHUNK) {
        const int pl = (np - pc < CHUNK) ? (np - pc) : CHUNK;
        for (int nc = 0; nc < nn; nc += CHUNK) {
            const int nl = (nn - nc < CHUNK) ? (nn - nc) : CHUNK;

            for (int i = tid; i < pl; i += 256) ldsP[i] = P[pc + i];
            for (int i = tid; i < nl; i += 256) ldsN[i] = Q[nc + i];
            __syncthreads();

            const int nPT = (pl + 15) >> 4;
            const int nNT = (nl + 15) >> 4;
            // uniform per-wave tile loop -> EXEC stays all-ones at the WMMA
            for (int t = wave; t < nPT * nNT; t += 8) {
                const int p0 = (t % nPT) << 4;
                const int n0 = (t / nPT) << 4;

                v2f A, B;
                if (lo) {
                    // lanes 0-15 carry A rows M=lane for K=0,1 and B cols
                    // N=lane for K=0,1 (ISA 32-bit 16x4 layout)
                    const int pi = p0 + lane;
                    const float sp = (pi < pl) ? ldsP[pi] : 0.0f;
                    A.x = -sp;  A.y = 1.0f;          // A[m,0]=-s_p, A[m,1]=1
                    const int ni = n0 + lane;
                    const float sn = (ni < nl) ? ldsN[ni] : 0.0f;
                    B.x = 1.0f; B.y = sn;            // B[0,n]=1,  B[1,n]=s_n
                } else {
                    // lanes 16-31 carry K=2,3 -> identically zero
                    A.x = 0.0f; A.y = 0.0f; B.x = 0.0f; B.y = 0.0f;
                }
                v8f C = {};
                // D[m,n] = s_n[n] - s_p[m]
                v8f D = __builtin_amdgcn_wmma_f32_16x16x4_f32(
                    /*neg_a=*/false, A, /*neg_b=*/false, B,
                    /*c_mod=*/(short)0, C, /*reuse_a=*/false, /*reuse_b=*/false);

                const int ncol   = lo ? lane : lane - 16;
                const bool nvOk  = (n0 + ncol) < nl;
                const int  mbase = lo ? 0 : 8;
#pragma unroll
                for (int v = 0; v < 8; ++v) {
                    const bool ok = nvOk && ((p0 + mbase + v) < pl);
                    const float x = D[v];
                    // softplus(x) = max(x,0) + ln(1 + exp(-|x|)) via raw
                    // hardware exp2/log2 (argument of log in (1,2]: safe)
                    const float ax = __builtin_fabsf(x);
                    const float e  = __builtin_amdgcn_exp2f(-ax * LOG2E);
                    const float l  = __builtin_amdgcn_logf(1.0f + e) * LN2;
                    const float sp = __builtin_fmaxf(x, 0.0f) + l;
                    acc += ok ? sp : 0.0f;
                }
            }
            __syncthreads();
        }
    }

    // deterministic reduction: fixed-order wave shuffle, then serial over waves
    for (int off = 16; off >= 1; off >>= 1)
        acc += __shfl_xor(acc, off, 32);
    if (lane == 0) waveSum[wave] = acc;
    __syncthreads();
    if (tid == 0) {
        float ssum = 0.0f;
        for (int w = 0; w < 8; ++w) ssum += waveSum[w];
        partial[g] = ssum;
    }
}

// ---------------------------------------------------------------------------
// Kernel 3: serial (deterministic) final reduction + mean.
// num_pairs = sum_g n_pos[g] * n_neg[g].
// ---------------------------------------------------------------------------
__global__ void finalize_kernel(const float* __restrict__ partial,
                                const unsigned* __restrict__ posCnt,
                                const unsigned* __restrict__ negCnt,
                                float* __restrict__ out)
{
    if (threadIdx.x == 0 && blockIdx.x == 0) {
        float ssum = 0.0f;
        unsigned long long pairs = 0ull;
        for (int g = 0; g < SEGS; ++g) {
            ssum  += partial[g];
            pairs += (unsigned long long)posCnt[g] * (unsigned long long)negCnt[g];
        }
        out[0] = ssum / (float)pairs;
    }
}

extern "C" void kernel_launch(void* const* d_in, const int* in_sizes, int n_in,
                              void* d_out, int out_size, void* d_ws, size_t ws_size,
                              hipStream_t stream) {
    const int*   b = (const int*)d_in[0];
    const float* s = (const float*)d_in[1];
    const int*   y = (const int*)d_in[2];
    const int    n = in_sizes[0];

    char* ws = (char*)d_ws;
    unsigned* posCnt  = (unsigned*)(ws + 0);
    unsigned* negCnt  = (unsigned*)(ws + 512);
    unsigned* posOff  = (unsigned*)(ws + 1024);
    unsigned* negOff  = (unsigned*)(ws + 1536);
    float*    partial = (float*)(ws + 2048);            // 128 floats
    float*    posScores = (float*)(ws + 4096);          // n floats
    float*    negScores = posScores + n;                // n floats

    bucket_kernel<<<1, 256, 0, stream>>>(b, s, y, n,
                                         posCnt, negCnt, posOff, negOff,
                                         posScores, negScores);
    pair_loss_kernel<<<SEGS, 256, 0, stream>>>(posScores, negScores,
                                               posCnt, negCnt, posOff, negOff,
                                               partial);
    finalize_kernel<<<1, 32, 0, stream>>>(partial, posCnt, negCnt, (float*)d_out);
}